// cg_interaction2_29480655519977
// MI455X (gfx1250) — compile-verified
//
#include <hip/hip_runtime.h>
#include <math.h>

// ---------------------------------------------------------------------------
// CG tensor-product kernel for MI455X (gfx1250, wave32).
//   out[e,b,c] = x + y + (TP21(xm,mid)[e,b] + TP22(ym,mid)[e,b])
//   mid = TP1(xm, ym),  xm/ym = channel means.
// TPs are evaluated as f32 WMMA GEMMs (V_WMMA_F32_16X16X4_F32):
//   GEMM1: M=16 edges, K=256 (i*16+j), N=32  (mid, k padded 25->32)
//   GEMM2/3: M=16 edges, K=400 (i*25+j), N=16
// W3J coefficients are computed on-device each launch (deterministic).
// ---------------------------------------------------------------------------

typedef float v2f __attribute__((ext_vector_type(2)));
typedef float v8f __attribute__((ext_vector_type(8)));

// workspace layout (float units)
#define W3J_OFF    0            // 100 slots * 729 floats (dense 9x9x9 per (a,b,c))
#define W3J_STRIDE 729
#define B1_OFF     72960        // float2[128][32]  (K=256 pairs x Nout32)
#define B2_OFF     81152        // float2[200][16]
#define B3_OFF     87552        // float2[200][16]
// total ws use: 93,952 floats = ~376 KB

__device__ __forceinline__ int slot_of(int a, int b, int c) { return (a * 5 + b) * 5 + c; }

// ---------------- su2 Clebsch-Gordan coefficient (double) -------------------
__device__ double cgcoef(int j1, int m1, int j2, int m2, int j3, int m3) {
    if (m1 + m2 != m3) return 0.0;
    const double f[14] = {1.0, 1.0, 2.0, 6.0, 24.0, 120.0, 720.0, 5040.0, 40320.0,
                          362880.0, 3628800.0, 39916800.0, 479001600.0, 6227020800.0};
    int vmin = max(0, max(j2 - j3 - m1, j1 - j3 + m2));
    int vmax = min(j1 + j2 - j3, min(j1 - m1, j2 + m2));
    double pref = sqrt((double)(2 * j3 + 1) * f[j3 + j1 - j2] * f[j3 - j1 + j2] *
                       f[j1 + j2 - j3] / f[j1 + j2 + j3 + 1]);
    pref *= sqrt(f[j3 + m3] * f[j3 - m3] * f[j1 + m1] * f[j1 - m1] * f[j2 + m2] * f[j2 - m2]);
    double s = 0.0;
    for (int v = vmin; v <= vmax; v++) {
        double d = f[v] * f[j1 + j2 - j3 - v] * f[j1 - m1 - v] * f[j2 + m2 - v] *
                   f[j3 - j2 + m1 + v] * f[j3 - j1 - m2 + v];
        s += ((v & 1) ? -1.0 : 1.0) / d;
    }
    return pref * s;
}

// q(l)[r][c] complex change-of-basis element, including (-i)^l factor.
__device__ __forceinline__ void qelem(int l, int r, int c, double& re, double& im) {
    const double irt2 = 0.70710678118654752440;
    double ar = 0.0, ai = 0.0;
    int m = r - l;
    if (m < 0) {
        if (c == l - m) ar = irt2;          // col l+|m|
        else if (c == l + m) ai = -irt2;    // col l-|m|
    } else if (m == 0) {
        if (c == l) ar = 1.0;
    } else {
        double sg = (m & 1) ? -1.0 : 1.0;
        if (c == l + m) ar = sg * irt2;
        else if (c == l - m) ai = sg * irt2;
    }
    switch (l & 3) {                        // multiply by (-i)^l
        case 0: re = ar;  im = ai;  break;
        case 1: re = ai;  im = -ar; break;
        case 2: re = -ar; im = -ai; break;
        default: re = -ai; im = ar; break;
    }
}

// ------------- kernel 1: dense real-basis W3J per (a,b,c) slot --------------
__global__ void w3j_kernel(float* __restrict__ ws) {
    int a = blockIdx.x / 25;
    int b = (blockIdx.x / 5) % 5;
    int c = blockIdx.x % 5;
    if (c < abs(a - b) || c > a + b || ((a + b + c) & 1)) return;
    const int da = 2 * a + 1, db = 2 * b + 1, dc = 2 * c + 1;
    const int n3 = da * db * dc;                    // <= 567
    __shared__ double bufR[576];
    __shared__ double bufA[576 * 2];
    __shared__ double bufB[576 * 2];
    __shared__ double red[256];
    int tid = threadIdx.x;

    for (int i = tid; i < n3; i += 256) bufR[i] = 0.0;
    __syncthreads();
    for (int ij = tid; ij < da * db; ij += 256) {   // su2 CG -> bufR
        int i = ij / db, j = ij % db;
        int m1 = i - a, m2 = j - b, m3 = m1 + m2;
        if (m3 >= -c && m3 <= c) bufR[ij * dc + (c + m3)] = cgcoef(a, m1, b, m2, c, m3);
    }
    __syncthreads();
    for (int idx = tid; idx < n3; idx += 256) {     // contract axis i with q(a)
        int jq = idx / (db * dc);
        int rem = idx - jq * (db * dc);
        double sre = 0.0, sim = 0.0;
        for (int i = 0; i < da; i++) {
            double qr, qi; qelem(a, i, jq, qr, qi);
            double cv = bufR[i * (db * dc) + rem];
            sre += qr * cv; sim += qi * cv;
        }
        bufA[2 * idx] = sre; bufA[2 * idx + 1] = sim;
    }
    __syncthreads();
    for (int idx = tid; idx < n3; idx += 256) {     // contract axis k with q(b)
        int jq = idx / (db * dc);
        int lq = (idx / dc) % db;
        int nn = idx % dc;
        double sre = 0.0, sim = 0.0;
        for (int k = 0; k < db; k++) {
            double qr, qi; qelem(b, k, lq, qr, qi);
            int src = (jq * db + k) * dc + nn;
            double ar = bufA[2 * src], ai = bufA[2 * src + 1];
            sre += qr * ar - qi * ai;
            sim += qr * ai + qi * ar;
        }
        bufB[2 * idx] = sre; bufB[2 * idx + 1] = sim;
    }
    __syncthreads();
    for (int idx = tid; idx < n3; idx += 256) {     // contract axis n with conj(q(c)[n][m]), take Re
        int pre = idx / dc;
        int m = idx % dc;
        double sre = 0.0;
        for (int nn = 0; nn < dc; nn++) {
            double qr, qi; qelem(c, nn, m, qr, qi);
            int src = pre * dc + nn;
            sre += qr * bufB[2 * src] + qi * bufB[2 * src + 1];   // Re(conj(q)*B)
        }
        bufR[idx] = sre;
    }
    __syncthreads();
    double s = 0.0;
    for (int idx = tid; idx < n3; idx += 256) s += bufR[idx] * bufR[idx];
    red[tid] = s; __syncthreads();
    for (int off = 128; off > 0; off >>= 1) {
        if (tid < off) red[tid] += red[tid + off];
        __syncthreads();
    }
    double inv = 1.0 / sqrt(red[0]);
    float* W = ws + W3J_OFF + (size_t)slot_of(a, b, c) * W3J_STRIDE;
    for (int idx = tid; idx < n3; idx += 256) {
        int i = idx / (db * dc);
        int j = (idx / dc) % db;
        int k = idx % dc;
        W[(i * 9 + j) * 9 + k] = (float)(bufR[idx] * inv);
    }
}

// ------------- kernel 2: fold w*pw into WMMA B-tables -----------------------
__global__ void build_kernel(const float* __restrict__ w1, const float* __restrict__ w21,
                             const float* __restrict__ w22, float* __restrict__ ws) {
    int tid = threadIdx.x;
    float* B1 = ws + B1_OFF;
    float* B2 = ws + B2_OFF;
    float* B3 = ws + B3_OFF;
    for (int i = tid; i < 128 * 32 * 2; i += 256) B1[i] = 0.f;
    for (int i = tid; i < 200 * 16 * 2; i += 256) { B2[i] = 0.f; B3[i] = 0.f; }
    __syncthreads();
    {   // TP1: (l1=3, l2=3, lmax_out=4)
        int cnt[5] = {0, 0, 0, 0, 0};
        for (int a = 0; a <= 3; a++) for (int b = 0; b <= 3; b++) for (int c = 0; c <= 4; c++)
            if (c >= abs(a - b) && c <= a + b && (((a + b + c) & 1) == 0)) cnt[c]++;
        int n = 0; unsigned counter = 0;
        for (int a = 0; a <= 3; a++) for (int b = 0; b <= 3; b++) for (int c = 0; c <= 4; c++) {
            if (!(c >= abs(a - b) && c <= a + b && (((a + b + c) & 1) == 0))) continue;
            float wv = w1[n] * sqrtf((float)(2 * c + 1) / (float)cnt[c]);
            const float* W = ws + W3J_OFF + (size_t)slot_of(a, b, c) * W3J_STRIDE;
            for (int ii = 0; ii < 2 * a + 1; ii++)
                for (int jj = 0; jj < 2 * b + 1; jj++)
                    for (int kk = 0; kk < 2 * c + 1; kk++) {
                        if ((counter++ & 255u) == (unsigned)tid) {
                            int K = (a * a + ii) * 16 + (b * b + jj);
                            int kO = c * c + kk;
                            B1[((K >> 1) * 32 + kO) * 2 + (K & 1)] += wv * W[(ii * 9 + jj) * 9 + kk];
                        }
                    }
            n++;
        }
    }
    {   // TP21 / TP22: (l1=3, l2=4, lmax_out=3), identical instruction lists
        int cnt[4] = {0, 0, 0, 0};
        for (int a = 0; a <= 3; a++) for (int b = 0; b <= 4; b++) for (int c = 0; c <= 3; c++)
            if (c >= abs(a - b) && c <= a + b && (((a + b + c) & 1) == 0)) cnt[c]++;
        int n = 0; unsigned counter = 0;
        for (int a = 0; a <= 3; a++) for (int b = 0; b <= 4; b++) for (int c = 0; c <= 3; c++) {
            if (!(c >= abs(a - b) && c <= a + b && (((a + b + c) & 1) == 0))) continue;
            float pw = sqrtf((float)(2 * c + 1) / (float)cnt[c]);
            float wa = w21[n] * pw, wb = w22[n] * pw;
            const float* W = ws + W3J_OFF + (size_t)slot_of(a, b, c) * W3J_STRIDE;
            for (int ii = 0; ii < 2 * a + 1; ii++)
                for (int jj = 0; jj < 2 * b + 1; jj++)
                    for (int kk = 0; kk < 2 * c + 1; kk++) {
                        if ((counter++ & 255u) == (unsigned)tid) {
                            int K = (a * a + ii) * 25 + (b * b + jj);
                            int kO = c * c + kk;
                            float v = W[(ii * 9 + jj) * 9 + kk];
                            B2[((K >> 1) * 16 + kO) * 2 + (K & 1)] += wa * v;
                            B3[((K >> 1) * 16 + kO) * 2 + (K & 1)] += wb * v;
                        }
                    }
            n++;
        }
    }
}

// ------------- kernel 3: main fused streaming + WMMA TP kernel --------------
// One wave32 per 16-edge tile. Lane l: h = l>>4 (K-half), n = l&15 (edge / N).
__global__ void __launch_bounds__(256) tp_kernel(const float* __restrict__ x,
                                                 const float* __restrict__ y,
                                                 const float* __restrict__ ws,
                                                 float* __restrict__ out, int ntiles) {
    const v2f* B1p = (const v2f*)(ws + B1_OFF);
    const v2f* B2p = (const v2f*)(ws + B2_OFF);
    const v2f* B3p = (const v2f*)(ws + B3_OFF);
    __shared__ float smem[8 * 1280];                 // per-wave scratch, 5KB each
    int tid = threadIdx.x;
    int wv = tid >> 5, l = tid & 31;
    int h = l >> 4, n = l & 15;
    float* xm_s = smem + wv * 1280;                  // [16][16]
    float* ym_s = xm_s + 256;                        // [16][16]
    float* mid_s = ym_s + 256;                       // [16][32]
    float* t_s = mid_s + 512;                        // [16][16]
    int gw = (blockIdx.x * blockDim.x + tid) >> 5;
    int nw = (gridDim.x * blockDim.x) >> 5;

    for (int tile = gw; tile < ntiles; tile += nw) {
        size_t ebase = (size_t)tile * 16;
        // ---- phase 1: channel means (coalesced), transpose via LDS ----
        for (int rr = 0; rr < 8; rr++) {
            int row = rr * 32 + l;                   // row = e*16 + b
            size_t f4 = ((ebase + (size_t)(row >> 4)) * 256 + (size_t)(row & 15) * 16) >> 2;
            const float4* px = ((const float4*)x) + f4;
            const float4* py = ((const float4*)y) + f4;
            float4 a0 = px[0], a1 = px[1], a2 = px[2], a3 = px[3];
            float xs = ((a0.x + a0.y) + (a0.z + a0.w)) + ((a1.x + a1.y) + (a1.z + a1.w)) +
                       ((a2.x + a2.y) + (a2.z + a2.w)) + ((a3.x + a3.y) + (a3.z + a3.w));
            float4 b0 = py[0], b1 = py[1], b2 = py[2], b3 = py[3];
            float ys = ((b0.x + b0.y) + (b0.z + b0.w)) + ((b1.x + b1.y) + (b1.z + b1.w)) +
                       ((b2.x + b2.y) + (b2.z + b2.w)) + ((b3.x + b3.y) + (b3.z + b3.w));
            xm_s[row] = xs * 0.0625f;
            ym_s[row] = ys * 0.0625f;
        }
        asm volatile("s_wait_dscnt 0" ::: "memory");
        float xr[16], yr[16];
        #pragma unroll
        for (int i = 0; i < 16; i++) { xr[i] = xm_s[n * 16 + i]; yr[i] = ym_s[n * 16 + i]; }

        // ---- phase 2: mid = GEMM1  (M=16 edges, K=256, N=32) ----
        v8f acc0 = {}, acc1 = {};
        #pragma unroll
        for (int s = 0; s < 64; s++) {
            const int K = 4 * s;
            float p0 = xr[(K + 0) >> 4] * yr[(K + 0) & 15];
            float p1 = xr[(K + 1) >> 4] * yr[(K + 1) & 15];
            float p2 = xr[(K + 2) >> 4] * yr[(K + 2) & 15];
            float p3 = xr[(K + 3) >> 4] * yr[(K + 3) & 15];
            v2f A = { h ? p2 : p0, h ? p3 : p1 };
            v2f b0 = B1p[(2 * s + h) * 32 + n];
            v2f b1 = B1p[(2 * s + h) * 32 + 16 + n];
            acc0 = __builtin_amdgcn_wmma_f32_16x16x4_f32(false, A, false, b0, (short)0, acc0, false, false);
            acc1 = __builtin_amdgcn_wmma_f32_16x16x4_f32(false, A, false, b1, (short)0, acc1, false, false);
        }
        #pragma unroll
        for (int r = 0; r < 8; r++) {                // D: M=r+8h (edge), N=n (k)
            mid_s[(r + 8 * h) * 32 + n] = acc0[r];
            mid_s[(r + 8 * h) * 32 + 16 + n] = acc1[r];
        }
        asm volatile("s_wait_dscnt 0" ::: "memory");
        float mr[25];
        #pragma unroll
        for (int j = 0; j < 25; j++) mr[j] = mid_s[n * 32 + j];

        // ---- phase 3: tp21/tp22 = GEMM2/3 (K=400, N=16) ----
        v8f acc2 = {}, acc3 = {};
        #pragma unroll
        for (int s = 0; s < 100; s++) {
            const int K = 4 * s;
            float q0 = xr[(K + 0) / 25] * mr[(K + 0) % 25];
            float q1 = xr[(K + 1) / 25] * mr[(K + 1) % 25];
            float q2 = xr[(K + 2) / 25] * mr[(K + 2) % 25];
            float q3 = xr[(K + 3) / 25] * mr[(K + 3) % 25];
            float r0 = yr[(K + 0) / 25] * mr[(K + 0) % 25];
            float r1 = yr[(K + 1) / 25] * mr[(K + 1) % 25];
            float r2 = yr[(K + 2) / 25] * mr[(K + 2) % 25];
            float r3 = yr[(K + 3) / 25] * mr[(K + 3) % 25];
            v2f A2 = { h ? q2 : q0, h ? q3 : q1 };
            v2f A3 = { h ? r2 : r0, h ? r3 : r1 };
            v2f b2 = B2p[(2 * s + h) * 16 + n];
            v2f b3 = B3p[(2 * s + h) * 16 + n];
            acc2 = __builtin_amdgcn_wmma_f32_16x16x4_f32(false, A2, false, b2, (short)0, acc2, false, false);
            acc3 = __builtin_amdgcn_wmma_f32_16x16x4_f32(false, A3, false, b3, (short)0, acc3, false, false);
        }
        #pragma unroll
        for (int r = 0; r < 8; r++) t_s[(r + 8 * h) * 16 + n] = acc2[r] + acc3[r];
        asm volatile("s_wait_dscnt 0" ::: "memory");

        // ---- phase 4: out = x + y + broadcast(t) (x/y re-read hit cache) ----
        for (int rr = 0; rr < 8; rr++) {
            int row = rr * 32 + l;                   // row = e*16 + b
            float tb = t_s[row];
            size_t f4 = ((ebase + (size_t)(row >> 4)) * 256 + (size_t)(row & 15) * 16) >> 2;
            const float4* px = ((const float4*)x) + f4;
            const float4* py = ((const float4*)y) + f4;
            float4* po = ((float4*)out) + f4;
            #pragma unroll
            for (int q = 0; q < 4; q++) {
                float4 a = px[q], b = py[q], o;
                o.x = a.x + b.x + tb; o.y = a.y + b.y + tb;
                o.z = a.z + b.z + tb; o.w = a.w + b.w + tb;
                po[q] = o;
            }
        }
    }
}

extern "C" void kernel_launch(void* const* d_in, const int* in_sizes, int n_in,
                              void* d_out, int out_size, void* d_ws, size_t ws_size,
                              hipStream_t stream) {
    const float* x = (const float*)d_in[0];
    const float* y = (const float*)d_in[1];
    const float* w1 = (const float*)d_in[2];
    const float* w21 = (const float*)d_in[3];
    const float* w22 = (const float*)d_in[4];
    float* ws = (float*)d_ws;
    float* out = (float*)d_out;
    int E = in_sizes[0] / 256;        // (E,16,16)
    int ntiles = E / 16;              // 12500 for E=200000

    hipLaunchKernelGGL(w3j_kernel, dim3(100), dim3(256), 0, stream, ws);
    hipLaunchKernelGGL(build_kernel, dim3(1), dim3(256), 0, stream, w1, w21, w22, ws);
    int blocks = (ntiles + 7) / 8;    // 1 tile per wave (8 waves/block)
    hipLaunchKernelGGL(tp_kernel, dim3(blocks), dim3(256), 0, stream, x, y, ws, out, ntiles);
}